// IncrementalRBFLayer_20212116095408
// MI455X (gfx1250) — compile-verified
//
#include <hip/hip_runtime.h>
#include <hip/hip_bf16.h>

typedef float v2f __attribute__((ext_vector_type(2)));
typedef float v8f __attribute__((ext_vector_type(8)));

#define KTOT 512          // centers
#define BTOT 8192         // batch
#define PROWS 272         // 256 quadratic terms + 16 linear terms
#define QROWS 136         // PROWS/2, pair-interleaved rows
// Pair layouts (row index p, pair q=p>>1, j=p&1):
//   Bmat[q*(2*KTOT) + k*2 + j] == Brow[p][k]      (Brow = [V_k flat ; -w_k])
//   XXp [q*(2*BTOT) + b*2 + j] == Arow[b][p]      (Arow = [x_d*x_e ; x_d])
// A lane (col,hf) then gets its WMMA fragment (K = 4c+2hf, 4c+2hf+1) with one b64 load.

// ---- prep 1: transpose vars (K,16,16) -> Bmat rows 0..255 ----
__global__ __launch_bounds__(256) void rbf_prep_transpose(const float* __restrict__ vars,
                                                          float* __restrict__ Bmat) {
  int idx = blockIdx.x * blockDim.x + threadIdx.x;   // 0 .. 256*KTOT-1
  if (idx >= 256 * KTOT) return;
  int p = idx >> 9;
  int k = idx & (KTOT - 1);
  Bmat[(p >> 1) * (2 * KTOT) + k * 2 + (p & 1)] = vars[k * 256 + p];
}

// ---- prep 2: w_k = (V+V^T) m_k -> rows 256..271 (negated); c_k = m^T V m ----
__global__ __launch_bounds__(256) void rbf_prep_wc(const float* __restrict__ vars,
                                                   const float* __restrict__ means,
                                                   float* __restrict__ Bmat,
                                                   float* __restrict__ cvec) {
  int k = blockIdx.x * blockDim.x + threadIdx.x;
  if (k >= KTOT) return;
  float m[16], w[16];
  #pragma unroll
  for (int e = 0; e < 16; ++e) m[e] = means[e * KTOT + k];   // means is (D,K)
  #pragma unroll
  for (int e = 0; e < 16; ++e) w[e] = 0.f;
  const float* V = vars + k * 256;
  float c = 0.f;
  #pragma unroll
  for (int d = 0; d < 16; ++d) {
    float rowacc = 0.f;
    #pragma unroll
    for (int e = 0; e < 16; ++e) {
      float v = V[d * 16 + e];
      rowacc += v * m[e];     // (V m)_d
      w[e]   += v * m[d];     // (V^T m)_e
    }
    w[d] += rowacc;
    c    += m[d] * rowacc;
  }
  #pragma unroll
  for (int d = 0; d < 16; ++d) {
    int p = 256 + d;
    Bmat[(p >> 1) * (2 * KTOT) + k * 2 + (p & 1)] = -w[d];
  }
  cvec[k] = c;
}

// ---- prep 3: A operand XX[b][p] = [x_d*x_e (p=d*16+e) ; x_d], pair layout ----
__global__ __launch_bounds__(256) void rbf_prep_xx(const float* __restrict__ x,
                                                   float* __restrict__ XXp) {
  int idx = blockIdx.x * blockDim.x + threadIdx.x;   // 0 .. QROWS*BTOT-1
  if (idx >= QROWS * BTOT) return;
  int q = idx >> 13;               // / BTOT
  int b = idx & (BTOT - 1);
  int p0 = 2 * q;                  // p0 even -> p0,p0+1 share d; never cross the 256 split
  float v0, v1;
  if (p0 < 256) {
    int d = p0 >> 4;
    int e = p0 & 15;
    float xd = x[b * 16 + d];
    v0 = xd * x[b * 16 + e];
    v1 = xd * x[b * 16 + e + 1];
  } else {
    v0 = x[b * 16 + (p0 - 256)];
    v1 = x[b * 16 + (p0 - 256) + 1];
  }
  v2f pr; pr[0] = v0; pr[1] = v1;
  *(v2f*)(XXp + (size_t)q * (2 * BTOT) + b * 2) = pr;
}

// ---- main GEMM: one wave per 16-batch x 64-center strip (4 k-tiles) ----
__global__ __launch_bounds__(32) void rbf_wmma_gemm(const float* __restrict__ XXp,
                                                    const float* __restrict__ Bmat,
                                                    const float* __restrict__ cvec,
                                                    float* __restrict__ out) {
  const int lane = threadIdx.x & 31;
  const int col  = lane & 15;      // N within tile / batch row for A
  const int hf   = lane >> 4;      // lane half selects K-pair
  const int bt   = blockIdx.x;     // batch tile 0..511
  const int kg   = blockIdx.y;     // k-group 0..7 (64 centers each)

  const float* ap = XXp + (size_t)hf * (2 * BTOT) + (bt * 16 + col) * 2;
  const float* bp = Bmat + (size_t)hf * (2 * KTOT) + (kg * 64 + col) * 2;

  v8f acc0 = {0.f,0.f,0.f,0.f,0.f,0.f,0.f,0.f};
  v8f acc1 = acc0, acc2 = acc0, acc3 = acc0;

  #pragma unroll 4
  for (int c = 0; c < 68; ++c) {
    v2f a  = *(const v2f*)(ap);
    v2f b0 = *(const v2f*)(bp +  0);
    v2f b1 = *(const v2f*)(bp + 32);
    v2f b2 = *(const v2f*)(bp + 64);
    v2f b3 = *(const v2f*)(bp + 96);
    acc0 = __builtin_amdgcn_wmma_f32_16x16x4_f32(false, a, false, b0, (short)0, acc0, false, false);
    acc1 = __builtin_amdgcn_wmma_f32_16x16x4_f32(false, a, false, b1, (short)0, acc1, false, false);
    acc2 = __builtin_amdgcn_wmma_f32_16x16x4_f32(false, a, false, b2, (short)0, acc2, false, false);
    acc3 = __builtin_amdgcn_wmma_f32_16x16x4_f32(false, a, false, b3, (short)0, acc3, false, false);
    ap += 4 * BTOT;    // next K-pair group
    bp += 4 * KTOT;
  }

  // epilogue: add c_k, exp(-0.5*q), store
  const int krow = kg * 64 + col;
  const float c0 = cvec[krow +  0];
  const float c1 = cvec[krow + 16];
  const float c2 = cvec[krow + 32];
  const float c3 = cvec[krow + 48];
  const int brow0 = bt * 16 + hf * 8;          // acc VGPR i holds M = i + 8*hf
  float* op = out + (size_t)brow0 * KTOT + krow;
  #pragma unroll
  for (int i = 0; i < 8; ++i) {
    op[i * KTOT +  0] = __expf(-0.5f * (acc0[i] + c0));
    op[i * KTOT + 16] = __expf(-0.5f * (acc1[i] + c1));
    op[i * KTOT + 32] = __expf(-0.5f * (acc2[i] + c2));
    op[i * KTOT + 48] = __expf(-0.5f * (acc3[i] + c3));
  }
}

extern "C" void kernel_launch(void* const* d_in, const int* in_sizes, int n_in,
                              void* d_out, int out_size, void* d_ws, size_t ws_size,
                              hipStream_t stream) {
  const float* x     = (const float*)d_in[0];
  const float* means = (const float*)d_in[1];
  const float* vars  = (const float*)d_in[2];
  float* out  = (float*)d_out;

  float* XXp  = (float*)d_ws;                          // QROWS*2*BTOT floats (8.9 MB)
  float* Bmat = XXp + (size_t)QROWS * (2 * BTOT);      // QROWS*2*KTOT floats (544 KB)
  float* cvec = Bmat + (size_t)QROWS * (2 * KTOT);     // 512 floats

  rbf_prep_transpose<<<(256 * KTOT + 255) / 256, 256, 0, stream>>>(vars, Bmat);
  rbf_prep_wc<<<(KTOT + 255) / 256, 256, 0, stream>>>(vars, means, Bmat, cvec);
  rbf_prep_xx<<<(QROWS * BTOT + 255) / 256, 256, 0, stream>>>(x, XXp);

  dim3 grid(BTOT / 16, KTOT / 64);                     // (512, 8) strips
  rbf_wmma_gemm<<<grid, 32, 0, stream>>>(XXp, Bmat, cvec, out);
}